// SelfObservingWrapper_45260365365709
// MI455X (gfx1250) — compile-verified
//
#include <hip/hip_runtime.h>
#include <cstddef>

#define B_  64
#define T_  1024
#define I_  256
#define H_  256
#define O_  256
#define DELAY_ 5
#define NWG 8
#define COLS 32   // output columns owned per workgroup

typedef float v2f __attribute__((ext_vector_type(2)));
typedef float v8f __attribute__((ext_vector_type(8)));

// ---- LDS layout (floats). Rows padded +4 floats: stride%64==4 -> no bank conflicts.
#define LWP   0
#define LWBB  (LWP  + 32*516)
#define LWF1  (LWBB + 32*516)
#define LWF2  (LWF1 + 32*260)
#define LWG   (LWF2 + 32*260)
#define LWH   (LWG  + 32*260)
#define LBIAS (LWH  + 32*260)      // 6 x 32: bp, bbb, bf1, bf2, bg, bh
#define LDS_FLOATS (LBIAS + 6*32)  // 66496 floats = 265984 B  (< 320 KB WGP LDS)

// ---- global workspace layout (floats)
#define WS_H    0
#define WS_PROJ (WS_H    + B_*H_)
#define WS_Z    (WS_PROJ + B_*H_)
#define WS_BUF  (WS_Z    + B_*H_)
#define WS_BAR  (WS_BUF  + DELAY_*B_*H_)   // 2 unsigned words (count, generation)
#define WS_ZERO_WORDS (WS_BAR + 2)

// ---------------------------------------------------------------------------
// fp32 WMMA K=256 accumulate: acc += A(16xK tile rows) x B(Kx16 from LDS W slice)
// A frag layout (ISA 7.12.2, 32-bit A 16x4): lanes 0-15 -> K={k,k+1}, lanes 16-31 -> K={k+2,k+3}
// B frag symmetric with N on lanes.
// ---------------------------------------------------------------------------
__device__ __forceinline__ v8f wmma_k256(v8f acc,
                                         const float* __restrict__ arow,  // A row for this lane
                                         const float* __restrict__ brow,  // LDS W row for this lane
                                         int half /* lane>>4 */)
{
    const int kb = half * 2;
#pragma unroll 8
    for (int k = 0; k < 256; k += 4) {
        v2f a = *(const v2f*)(arow + k + kb);
        v2f b = *(const v2f*)(brow + k + kb);
        acc = __builtin_amdgcn_wmma_f32_16x16x4_f32(false, a, false, b,
                                                    (short)0, acc, false, false);
    }
    return acc;
}

// ---------------------------------------------------------------------------
// device-scope grid barrier over NWG resident workgroups.
// thread 0 signals; lane 0 of EVERY wave performs the agent-scope acquire spin
// so each wave's cache path observes remote writes before proceeding.
// ---------------------------------------------------------------------------
__device__ __forceinline__ void grid_barrier(unsigned* cnt, unsigned* gen, unsigned& mygen)
{
    __syncthreads();
    if (threadIdx.x == 0) {
        unsigned prev = __hip_atomic_fetch_add(cnt, 1u, __ATOMIC_ACQ_REL,
                                               __HIP_MEMORY_SCOPE_AGENT);
        if (prev == NWG - 1u) {
            __hip_atomic_store(cnt, 0u, __ATOMIC_RELAXED, __HIP_MEMORY_SCOPE_AGENT);
            __hip_atomic_fetch_add(gen, 1u, __ATOMIC_RELEASE, __HIP_MEMORY_SCOPE_AGENT);
        }
    }
    if ((threadIdx.x & 31) == 0) {
        while (__hip_atomic_load(gen, __ATOMIC_ACQUIRE,
                                 __HIP_MEMORY_SCOPE_AGENT) == mygen) {
            __builtin_amdgcn_s_sleep(1);
        }
    }
    mygen++;
    __syncthreads();
}

__global__ void init_ws_kernel(float* __restrict__ ws)
{
    size_t i = (size_t)blockIdx.x * blockDim.x + threadIdx.x;
    if (i < (size_t)WS_ZERO_WORDS) ws[i] = 0.0f;   // zeros h, ring buffer, barrier words
}

// ---------------------------------------------------------------------------
// Persistent CfC kernel: 8 WGs x 256 threads; weights resident in LDS.
// ---------------------------------------------------------------------------
__global__ void __launch_bounds__(256) cfc_persistent_kernel(
    const float* __restrict__ x,
    const float* __restrict__ Wp,  const float* __restrict__ bp,
    const float* __restrict__ Wbb, const float* __restrict__ bbb,
    const float* __restrict__ Wf1, const float* __restrict__ bf1,
    const float* __restrict__ Wf2, const float* __restrict__ bf2,
    const float* __restrict__ Wta, const float* __restrict__ bta,
    const float* __restrict__ Wtb, const float* __restrict__ btb,
    const float* __restrict__ Wh,  const float* __restrict__ bh,
    float* __restrict__ preds, float* __restrict__ hiddens, float* __restrict__ ws)
{
    extern __shared__ float lds[];
    const int tid  = threadIdx.x;
    const int wg   = blockIdx.x;
    const int lane = tid & 31;
    const int wave = tid >> 5;
    const int mt   = wave >> 1;           // 4 M tiles of 16 rows (batch)
    const int nt   = wave & 1;            // 2 N tiles of 16 cols
    const int l16  = lane & 15;
    const int half = lane >> 4;
    const int ncol_local = nt * 16 + l16;        // 0..31 within slice
    const int ncol       = wg * COLS + ncol_local;

    // ---- one-time: load this WG's weight column-slice into LDS (fp32) ----
    for (int idx = tid; idx < 32 * 512; idx += 256) {
        int r = idx >> 9, c = idx & 511;
        size_t g = (size_t)(wg * 32 + r) * 512 + c;
        lds[LWP  + r * 516 + c] = Wp[g];
        lds[LWBB + r * 516 + c] = Wbb[g];
    }
    for (int idx = tid; idx < 32 * 256; idx += 256) {
        int r = idx >> 8, c = idx & 255;
        size_t g = (size_t)(wg * 32 + r) * 256 + c;
        lds[LWF1 + r * 260 + c] = Wf1[g];
        lds[LWF2 + r * 260 + c] = Wf2[g];
        lds[LWG  + r * 260 + c] = Wta[g] + Wtb[g];   // fold the two gate GEMMs
        lds[LWH  + r * 260 + c] = Wh[g];
    }
    if (tid < 32) {
        int n = wg * 32 + tid;
        lds[LBIAS + 0 * 32 + tid] = bp[n];
        lds[LBIAS + 1 * 32 + tid] = bbb[n];
        lds[LBIAS + 2 * 32 + tid] = bf1[n];
        lds[LBIAS + 3 * 32 + tid] = bf2[n];
        lds[LBIAS + 4 * 32 + tid] = bta[n] + btb[n];
        lds[LBIAS + 5 * 32 + tid] = bh[n];
    }
    __syncthreads();

    float*    hbuf = ws + WS_H;
    float*    proj = ws + WS_PROJ;
    float*    zbuf = ws + WS_Z;
    float*    ring = ws + WS_BUF;
    unsigned* bcnt = (unsigned*)(ws + WS_BAR);
    unsigned* bgen = bcnt + 1;
    unsigned  mygen = 0;

    const size_t ldX    = (size_t)T_ * I_;
    const int    arow_m = mt * 16 + l16;

    const float* brow_wp_x = &lds[LWP  + ncol_local * 516 + 0];
    const float* brow_wp_h = &lds[LWP  + ncol_local * 516 + 256];
    const float* brow_bb_p = &lds[LWBB + ncol_local * 516 + 0];
    const float* brow_bb_h = &lds[LWBB + ncol_local * 516 + 256];
    const float* brow_f1   = &lds[LWF1 + ncol_local * 260];
    const float* brow_f2   = &lds[LWF2 + ncol_local * 260];
    const float* brow_g    = &lds[LWG  + ncol_local * 260];
    const float* brow_o    = &lds[LWH  + ncol_local * 260];

    const float* arow_p = proj + (size_t)arow_m * H_;
    const float* arow_h = hbuf + (size_t)arow_m * H_;
    const float* arow_z = zbuf + (size_t)arow_m * H_;
    const int    mbase  = mt * 16 + half * 8;

    for (int t = 0; t < T_; ++t) {
        const int slot = t % DELAY_;

        // ---- phase 1: proj = x_t @ WpX^T + delayed @ WpH^T + bp ----
        const float* arow_x = x + (size_t)arow_m * ldX + (size_t)t * I_;
        const float* arow_d = ring + (size_t)slot * (B_ * H_) + (size_t)arow_m * H_;
        v8f acc = {};
        acc = wmma_k256(acc, arow_x, brow_wp_x, half);
        acc = wmma_k256(acc, arow_d, brow_wp_h, half);
        {
            const float b = lds[LBIAS + 0 * 32 + ncol_local];
#pragma unroll
            for (int v = 0; v < 8; ++v)
                proj[(size_t)(mbase + v) * H_ + ncol] = acc[v] + b;
        }
        grid_barrier(bcnt, bgen, mygen);

        // ---- phase 2: z = lecun_tanh(proj @ WbbP^T + h @ WbbH^T + bbb) ----
        v8f az = {};
        az = wmma_k256(az, arow_p, brow_bb_p, half);
        az = wmma_k256(az, arow_h, brow_bb_h, half);
        {
            const float b = lds[LBIAS + 1 * 32 + ncol_local];
#pragma unroll
            for (int v = 0; v < 8; ++v)
                zbuf[(size_t)(mbase + v) * H_ + ncol] =
                    1.7159f * tanhf(0.666f * (az[v] + b));
        }
        grid_barrier(bcnt, bgen, mygen);

        // ---- phase 3: ff1/ff2/gate -> h_new ----
        v8f u1 = {}, u2 = {}, ug = {};
        u1 = wmma_k256(u1, arow_z, brow_f1, half);
        u2 = wmma_k256(u2, arow_z, brow_f2, half);
        ug = wmma_k256(ug, arow_z, brow_g,  half);
        {
            const float b1 = lds[LBIAS + 2 * 32 + ncol_local];
            const float b2 = lds[LBIAS + 3 * 32 + ncol_local];
            const float bg = lds[LBIAS + 4 * 32 + ncol_local];
#pragma unroll
            for (int v = 0; v < 8; ++v) {
                float f1 = tanhf(u1[v] + b1);
                float f2 = tanhf(u2[v] + b2);
                float s  = 1.0f / (1.0f + expf(-(ug[v] + bg)));
                float hn = f1 * (1.0f - s) + s * f2;
                int m = mbase + v;
                hbuf[(size_t)m * H_ + ncol] = hn;
                ring[(size_t)slot * (B_ * H_) + (size_t)m * H_ + ncol] = hn;
                hiddens[((size_t)m * T_ + t) * H_ + ncol] = hn;
            }
        }
        grid_barrier(bcnt, bgen, mygen);

        // ---- phase 4: pred = h_new @ Wh^T + bh (overlaps next step's phase 1) ----
        // Prefetch next timestep's x A-rows (global_prefetch_b8) to hide the only
        // HBM-sourced operand of the next iteration's dependency chain.
        if (t + 1 < T_) {
            const float* nx = x + (size_t)arow_m * ldX + (size_t)(t + 1) * I_;
            __builtin_prefetch(nx + (half * 128), 0, 3);
        }
        v8f up = {};
        up = wmma_k256(up, arow_h, brow_o, half);
        {
            const float b = lds[LBIAS + 5 * 32 + ncol_local];
#pragma unroll
            for (int v = 0; v < 8; ++v)
                preds[((size_t)(mbase + v) * T_ + t) * O_ + ncol] = up[v] + b;
        }
    }
}

extern "C" void kernel_launch(void* const* d_in, const int* in_sizes, int n_in,
                              void* d_out, int out_size, void* d_ws, size_t ws_size,
                              hipStream_t stream)
{
    (void)in_sizes; (void)n_in; (void)out_size; (void)ws_size;
    const float* x   = (const float*)d_in[0];
    const float* Wp  = (const float*)d_in[1];
    const float* bp  = (const float*)d_in[2];
    const float* Wbb = (const float*)d_in[3];
    const float* bbb = (const float*)d_in[4];
    const float* Wf1 = (const float*)d_in[5];
    const float* bf1 = (const float*)d_in[6];
    const float* Wf2 = (const float*)d_in[7];
    const float* bf2 = (const float*)d_in[8];
    const float* Wta = (const float*)d_in[9];
    const float* bta = (const float*)d_in[10];
    const float* Wtb = (const float*)d_in[11];
    const float* btb = (const float*)d_in[12];
    const float* Wh  = (const float*)d_in[13];
    const float* bh  = (const float*)d_in[14];

    float* preds   = (float*)d_out;
    float* hiddens = preds + (size_t)B_ * T_ * O_;
    float* ws      = (float*)d_ws;

    // allow >64KB dynamic LDS (266KB slice per WGP); idempotent, not a stream op
    (void)hipFuncSetAttribute((const void*)cfc_persistent_kernel,
                              hipFuncAttributeMaxDynamicSharedMemorySize,
                              (int)(LDS_FLOATS * sizeof(float)));

    const int words = WS_ZERO_WORDS;
    init_ws_kernel<<<(words + 255) / 256, 256, 0, stream>>>(ws);

    cfc_persistent_kernel<<<dim3(NWG), dim3(256),
                            LDS_FLOATS * sizeof(float), stream>>>(
        x, Wp, bp, Wbb, bbb, Wf1, bf1, Wf2, bf2,
        Wta, bta, Wtb, btb, Wh, bh, preds, hiddens, ws);
}